// PtModule_73667279061103
// MI455X (gfx1250) — compile-verified
//
#include <hip/hip_runtime.h>

// y = 2*x + 3 over 2^26 fp32 elements. Pure HBM stream: 512 MiB traffic,
// 0.25 flop/byte -> bound by 23.3 TB/s (floor ~22 us). Optimal MI455X shape:
//   - b128 global loads/stores with TH=NT (512 MiB stream vs 192 MB L2)
//   - branch-free hot kernel: 4 loads claused up front, staggered waits
//   - 32-bit indexing -> saddr+voffset addressing, minimal SALU/VALU overhead
//   - 256-thread blocks = 8 wave32 waves; 16384 blocks of concurrency
// No WMMA on purpose: matrix ops cannot help an op ~1000x below machine balance.

typedef float v4f __attribute__((ext_vector_type(4)));

#define BLOCK_THREADS 256u
#define V4_PER_THREAD 4u                                   // 16 floats/thread
#define FLOATS_PER_BLOCK (BLOCK_THREADS * V4_PER_THREAD * 4u)  // 4096

static __device__ __forceinline__ v4f affine(v4f v) {
    v4f r;
    r.x = __builtin_fmaf(v.x, 2.0f, 3.0f);
    r.y = __builtin_fmaf(v.y, 2.0f, 3.0f);
    r.z = __builtin_fmaf(v.z, 2.0f, 3.0f);
    r.w = __builtin_fmaf(v.w, 2.0f, 3.0f);
    return r;
}

// Hot kernel: grid exactly covers full blocks -> no bounds checks anywhere.
__global__ __launch_bounds__(BLOCK_THREADS) void affine2x3_main(const float* __restrict__ x,
                                                                float* __restrict__ y) {
    const v4f* __restrict__ xv = (const v4f*)x;
    v4f* __restrict__       yv = (v4f*)y;

    // 32-bit float4 indices (max 2^24 here; byte offset < 2^28).
    const unsigned base = blockIdx.x * (BLOCK_THREADS * V4_PER_THREAD) + threadIdx.x;
    const unsigned i0 = base;
    const unsigned i1 = base + BLOCK_THREADS;
    const unsigned i2 = base + 2u * BLOCK_THREADS;
    const unsigned i3 = base + 3u * BLOCK_THREADS;

    // Issue all loads before any consumer: one s_clause of 4 b128 NT loads,
    // waits staggered as data returns.
    v4f a0 = __builtin_nontemporal_load(xv + i0);
    v4f a1 = __builtin_nontemporal_load(xv + i1);
    v4f a2 = __builtin_nontemporal_load(xv + i2);
    v4f a3 = __builtin_nontemporal_load(xv + i3);

    a0 = affine(a0);
    a1 = affine(a1);
    a2 = affine(a2);
    a3 = affine(a3);

    __builtin_nontemporal_store(a0, yv + i0);
    __builtin_nontemporal_store(a1, yv + i1);
    __builtin_nontemporal_store(a2, yv + i2);
    __builtin_nontemporal_store(a3, yv + i3);
}

// Tail kernel: only launched if n % FLOATS_PER_BLOCK != 0 (not for this shape).
__global__ __launch_bounds__(BLOCK_THREADS) void affine2x3_tail(const float* __restrict__ x,
                                                                float* __restrict__ y,
                                                                long long start,
                                                                long long n) {
    long long t = start + (long long)blockIdx.x * BLOCK_THREADS + threadIdx.x;
    if (t < n) {
        y[t] = __builtin_fmaf(x[t], 2.0f, 3.0f);
    }
}

extern "C" void kernel_launch(void* const* d_in, const int* in_sizes, int n_in,
                              void* d_out, int out_size, void* d_ws, size_t ws_size,
                              hipStream_t stream) {
    (void)n_in; (void)d_ws; (void)ws_size; (void)out_size;

    const float* x = (const float*)d_in[0];
    float*       y = (float*)d_out;
    const long long n = (long long)in_sizes[0];   // 67,108,864

    const long long full_blocks = n / FLOATS_PER_BLOCK;   // 16,384 for this shape
    if (full_blocks > 0) {
        affine2x3_main<<<(unsigned)full_blocks, BLOCK_THREADS, 0, stream>>>(x, y);
    }

    const long long start = full_blocks * (long long)FLOATS_PER_BLOCK;
    const long long rem   = n - start;                    // 0 for this shape
    if (rem > 0) {
        const unsigned tail_grid = (unsigned)((rem + BLOCK_THREADS - 1) / BLOCK_THREADS);
        affine2x3_tail<<<tail_grid, BLOCK_THREADS, 0, stream>>>(x, y, start, n);
    }
}